// GCN_32066225832361
// MI455X (gfx1250) — compile-verified
//
#include <hip/hip_runtime.h>

typedef float v2f __attribute__((ext_vector_type(2)));
typedef float v8f __attribute__((ext_vector_type(8)));

#define IN_F 256
#define HID 64
#define OUTF 32

// ---------------------------------------------------------------------------
// WMMA fp32 GEMM:  C[N x NCOL] = A[N x K] * B[K x NCOL]   (row-major all)
// One wave computes a 16-row tile across all NCOL columns using
// V_WMMA_F32_16X16X4_F32.  A-fragment layout (ISA 7.12.2, 32-bit A 16x4):
//   lane m=lane&15 -> row M=m ; g=lane>>4 -> holds K = {2g, 2g+1}
// B (4x16) mirrored: lane holds B[2g + {0,1}][n=m].
// C/D: VGPR v -> row (v + 8g), col n=m.
// ---------------------------------------------------------------------------
template <int K, int NCOL>
__global__ void gcn_gemm_wmma(const float* __restrict__ A,
                              const float* __restrict__ B,
                              float* __restrict__ C, int nTiles) {
  constexpr int NT = NCOL / 16;
  const int lane = threadIdx.x & 31;
  const int wave = threadIdx.x >> 5;
  const int tile = blockIdx.x * (blockDim.x >> 5) + wave;
  if (tile >= nTiles) return;  // wave-uniform: surviving waves keep EXEC==~0

  const int m = lane & 15;
  const int g = lane >> 4;
  const float* arow = A + ((long)tile * 16 + m) * K + 2 * g;

  v8f acc[NT];
#pragma unroll
  for (int t = 0; t < NT; ++t) acc[t] = (v8f){0.f,0.f,0.f,0.f,0.f,0.f,0.f,0.f};

  for (int kb = 0; kb < K; kb += 4) {
    const v2f a = *(const v2f*)(arow + kb);  // A[row][kb+2g], A[row][kb+2g+1]
#pragma unroll
    for (int t = 0; t < NT; ++t) {
      v2f b;
      const int col = t * 16 + m;
      b.x = B[(kb + 2 * g) * NCOL + col];
      b.y = B[(kb + 2 * g + 1) * NCOL + col];
      acc[t] = __builtin_amdgcn_wmma_f32_16x16x4_f32(
          false, a, false, b, (short)0, acc[t], false, false);
    }
  }

  const long rowBase = (long)tile * 16;
#pragma unroll
  for (int v = 0; v < 8; ++v) {
    const long r = rowBase + v + 8 * g;
#pragma unroll
    for (int t = 0; t < NT; ++t) C[r * NCOL + t * 16 + m] = acc[t][v];
  }
}

// ---------------------------------------------------------------------------
__global__ void k_fill(float* __restrict__ p, float v, int n) {
  int i = blockIdx.x * blockDim.x + threadIdx.x;
  if (i < n) p[i] = v;
}

__global__ void k_deg(const int* __restrict__ ei, const float* __restrict__ ew,
                      float* __restrict__ deg, int E) {
  int e = blockIdx.x * blockDim.x + threadIdx.x;
  if (e < E) atomicAdd(&deg[ei[E + e]], ew[e]);  // dst row of edge_index
}

__global__ void k_dinv(const float* __restrict__ deg, float* __restrict__ dinv,
                       int n) {
  int i = blockIdx.x * blockDim.x + threadIdx.x;
  if (i < n) dinv[i] = rsqrtf(deg[i]);
}

__global__ void k_norm(const int* __restrict__ ei, const float* __restrict__ ew,
                       const float* __restrict__ dinv, float* __restrict__ nrm,
                       int E) {
  int e = blockIdx.x * blockDim.x + threadIdx.x;
  if (e < E) nrm[e] = dinv[ei[e]] * ew[e] * dinv[ei[E + e]];
}

// agg[i][k] = dinv[i]^2 * h[i][k] + bias[k]
__global__ void k_selfinit(const float* __restrict__ h,
                           const float* __restrict__ dinv,
                           const float* __restrict__ bias,
                           float* __restrict__ agg, long total, int C) {
  long idx = (long)blockIdx.x * blockDim.x + threadIdx.x;
  if (idx >= total) return;
  long i = idx / C;
  int k = (int)(idx % C);
  float d = dinv[i];
  agg[idx] = d * d * h[idx] + bias[k];
}

// one thread handles 4 features of one edge: agg[dst] += norm * h[src]
template <int C>
__global__ void k_scatter4(const int* __restrict__ ei,
                           const float* __restrict__ nrm,
                           const float* __restrict__ h,
                           float* __restrict__ agg, int E) {
  constexpr int PER = C / 4;
  long t = (long)blockIdx.x * blockDim.x + threadIdx.x;
  long e = t / PER;
  if (e >= E) return;
  int k4 = (int)(t % PER) * 4;
  int src = ei[e];
  int dst = ei[E + e];
  float w = nrm[e];
  const float4 v = *(const float4*)(h + (long)src * C + k4);
  float* o = agg + (long)dst * C + k4;
  atomicAdd(o + 0, w * v.x);
  atomicAdd(o + 1, w * v.y);
  atomicAdd(o + 2, w * v.z);
  atomicAdd(o + 3, w * v.w);
}

__global__ void k_relu(float* __restrict__ p, long n) {
  long i = (long)blockIdx.x * blockDim.x + threadIdx.x;
  if (i < n) p[i] = fmaxf(p[i], 0.f);
}

// attention coefs: coef[3][N] written to output tail
__global__ void k_coef(const float* __restrict__ e1, const float* __restrict__ e2,
                       const float* __restrict__ e3, const float* __restrict__ fw,
                       const float* __restrict__ fb, float* __restrict__ coef,
                       int n) {
  int i = blockIdx.x * blockDim.x + threadIdx.x;
  if (i >= n) return;
  float d1 = fb[0], d2 = fb[0], d3 = fb[0];
  const float* p1 = e1 + (long)i * HID;
  const float* p2 = e2 + (long)i * HID;
  const float* p3 = e3 + (long)i * HID;
#pragma unroll 8
  for (int k = 0; k < HID; ++k) {
    float w = fw[k];
    d1 += p1[k] * w;
    d2 += p2[k] * w;
    d3 += p3[k] * w;
  }
  float c1 = expf(d1 > 0.f ? d1 : 0.01f * d1);
  float c2 = expf(d2 > 0.f ? d2 : 0.01f * d2);
  float c3 = expf(d3 > 0.f ? d3 : 0.01f * d3);
  float inv = 1.f / (c1 + c2 + c3);
  coef[i] = c1 * inv;
  coef[n + i] = c2 * inv;
  coef[2 * n + i] = c3 * inv;
}

// combined[i][k] = sum_b e_b[i][k] * coef_b[i]   (thread = node x 4 feats)
__global__ void k_combine(const float* __restrict__ e1, const float* __restrict__ e2,
                          const float* __restrict__ e3,
                          const float* __restrict__ coef,
                          float* __restrict__ comb, int n) {
  long idx = (long)blockIdx.x * blockDim.x + threadIdx.x;
  if (idx >= (long)n * (HID / 4)) return;
  long i = idx / (HID / 4);
  int k4 = (int)(idx % (HID / 4)) * 4;
  float c1 = coef[i], c2 = coef[n + i], c3 = coef[2 * n + i];
  const float4 a = *(const float4*)(e1 + i * HID + k4);
  const float4 b = *(const float4*)(e2 + i * HID + k4);
  const float4 c = *(const float4*)(e3 + i * HID + k4);
  float4 r;
  r.x = a.x * c1 + b.x * c2 + c.x * c3;
  r.y = a.y * c1 + b.y * c2 + c.y * c3;
  r.z = a.z * c1 + b.z * c2 + c.z * c3;
  r.w = a.w * c1 + b.w * c2 + c.w * c3;
  *(float4*)(comb + i * HID + k4) = r;
}

// out[i][k] = sum_b ( dinv_b[i]^2 * h2_b[i][k] + bias_b[k] )
__global__ void k_outinit(const float* __restrict__ h2a, const float* __restrict__ h2b,
                          const float* __restrict__ h2c,
                          const float* __restrict__ d1, const float* __restrict__ d2,
                          const float* __restrict__ d3,
                          const float* __restrict__ ba, const float* __restrict__ bb,
                          const float* __restrict__ bc,
                          float* __restrict__ out, long total) {
  long idx = (long)blockIdx.x * blockDim.x + threadIdx.x;
  if (idx >= total) return;
  long i = idx / OUTF;
  int k = (int)(idx % OUTF);
  float a = d1[i], b = d2[i], c = d3[i];
  out[idx] = a * a * h2a[idx] + b * b * h2b[idx] + c * c * h2c[idx] +
             ba[k] + bb[k] + bc[k];
}

// ---------------------------------------------------------------------------
extern "C" void kernel_launch(void* const* d_in, const int* in_sizes, int n_in,
                              void* d_out, int out_size, void* d_ws,
                              size_t ws_size, hipStream_t stream) {
  const float* x[3]  = {(const float*)d_in[0], (const float*)d_in[1], (const float*)d_in[2]};
  const int*   ei[3] = {(const int*)d_in[3], (const int*)d_in[4], (const int*)d_in[5]};
  const float* ew[3] = {(const float*)d_in[6], (const float*)d_in[7], (const float*)d_in[8]};
  const float* W1[3] = {(const float*)d_in[9], (const float*)d_in[10], (const float*)d_in[11]};
  const float* b1[3] = {(const float*)d_in[12], (const float*)d_in[13], (const float*)d_in[14]};
  const float* fc_w  = (const float*)d_in[15];
  const float* fc_b  = (const float*)d_in[16];
  const float* W2[3] = {(const float*)d_in[17], (const float*)d_in[18], (const float*)d_in[19]};
  const float* b2[3] = {(const float*)d_in[20], (const float*)d_in[21], (const float*)d_in[22]};

  const int N = in_sizes[0] / IN_F;   // 50000
  const int E = in_sizes[3] / 2;      // 800000
  float* out = (float*)d_out;
  float* coefs = out + (long)N * OUTF;  // output tail: coef1|coef2|coef3

  // workspace carve-up (floats)
  float* ws = (float*)d_ws;
  long o = 0;
  float* h = ws + o;       o += (long)N * HID;      // gemm1 scratch (reused)
  float* eb[3];
  for (int b = 0; b < 3; ++b) { eb[b] = ws + o; o += (long)N * HID; }
  float* comb = ws + o;    o += (long)N * HID;
  float* h2[3];
  for (int b = 0; b < 3; ++b) { h2[b] = ws + o; o += (long)N * OUTF; }
  float* dinv[3];
  for (int b = 0; b < 3; ++b) { dinv[b] = ws + o; o += N; }
  float* deg = ws + o;     o += N;
  float* nrm[3];
  for (int b = 0; b < 3; ++b) { nrm[b] = ws + o; o += E; }

  const int BLK = 256;
  auto cdiv = [](long a, long b) { return (int)((a + b - 1) / b); };
  const int nTiles = N / 16;                 // N divisible by 16
  const int gemmGrid = cdiv(nTiles, BLK / 32);

  // ---------------- layer 1 (per branch) ----------------
  for (int b = 0; b < 3; ++b) {
    gcn_gemm_wmma<IN_F, HID><<<gemmGrid, BLK, 0, stream>>>(x[b], W1[b], h, nTiles);
    k_fill<<<cdiv(N, BLK), BLK, 0, stream>>>(deg, 1.0f, N);
    k_deg<<<cdiv(E, BLK), BLK, 0, stream>>>(ei[b], ew[b], deg, E);
    k_dinv<<<cdiv(N, BLK), BLK, 0, stream>>>(deg, dinv[b], N);
    k_norm<<<cdiv(E, BLK), BLK, 0, stream>>>(ei[b], ew[b], dinv[b], nrm[b], E);
    k_selfinit<<<cdiv((long)N * HID, BLK), BLK, 0, stream>>>(h, dinv[b], b1[b],
                                                             eb[b], (long)N * HID, HID);
    k_scatter4<HID><<<cdiv((long)E * (HID / 4), BLK), BLK, 0, stream>>>(
        ei[b], nrm[b], h, eb[b], E);
    k_relu<<<cdiv((long)N * HID, BLK), BLK, 0, stream>>>(eb[b], (long)N * HID);
  }

  // ---------------- attention + combine ----------------
  k_coef<<<cdiv(N, BLK), BLK, 0, stream>>>(eb[0], eb[1], eb[2], fc_w, fc_b, coefs, N);
  k_combine<<<cdiv((long)N * (HID / 4), BLK), BLK, 0, stream>>>(eb[0], eb[1], eb[2],
                                                                coefs, comb, N);

  // ---------------- layer 2 ----------------
  for (int b = 0; b < 3; ++b)
    gcn_gemm_wmma<HID, OUTF><<<gemmGrid, BLK, 0, stream>>>(comb, W2[b], h2[b], nTiles);

  k_outinit<<<cdiv((long)N * OUTF, BLK), BLK, 0, stream>>>(
      h2[0], h2[1], h2[2], dinv[0], dinv[1], dinv[2], b2[0], b2[1], b2[2], out,
      (long)N * OUTF);

  for (int b = 0; b < 3; ++b)
    k_scatter4<OUTF><<<cdiv((long)E * (OUTF / 4), BLK), BLK, 0, stream>>>(
        ei[b], nrm[b], h2[b], out, E);
}